// FocalLoss_10754598109558
// MI455X (gfx1250) — compile-verified
//
#include <hip/hip_runtime.h>

// Problem constants (from reference): B=8, C=21, H=W=512
#define HWSZ   262144          // 512*512
#define NCH    21
#define CHW    (NCH * HWSZ)    // 5,505,024 floats per batch image
#define NBATCH 8
#define NPIX   (NBATCH * HWSZ) // 2,097,152 pixels total
#define NBLK   1024
#define NTHR   256

typedef __attribute__((ext_vector_type(2))) float v2f;
typedef __attribute__((ext_vector_type(8))) float v8f;

// Stage 1: per-block partial sums of focal loss. Thread g handles pixel g of
// each of the 8 batch images -> per-channel loads are 128B-contiguous per wave.
__global__ __launch_bounds__(NTHR) void focal_partial_kernel(
    const float* __restrict__ inp,   // [B, C, H, W] f32
    const int*   __restrict__ tgt,   // [B, H, W] int (values 0..20 or 255)
    float*       __restrict__ part)  // [NBLK] partial sums
{
    const int g = blockIdx.x * NTHR + threadIdx.x;     // pixel index within one image
    const float LOG2E = 1.4426950408889634f;
    const float LN2   = 0.6931471805599453f;

    float acc = 0.0f;

    #pragma unroll 1
    for (int b = 0; b < NBATCH; ++b) {
        const float* px = inp + (size_t)b * CHW + g;
        if (b + 1 < NBATCH) {
            // CDNA5 global_prefetch_b8: pull next batch image's line toward L2
            __builtin_prefetch(px + CHW, 0, 0);
        }

        // Pass 1: load all 21 channel values into registers, track max
        float v[NCH];
        float m = -3.402823466e38f;
        #pragma unroll
        for (int c = 0; c < NCH; ++c) {
            v[c] = px[(size_t)c * HWSZ];
            m = fmaxf(m, v[c]);
        }

        int t = tgt[b * HWSZ + g];
        t = (t == 255) ? 0 : t;   // remap 255 -> 0 (mask is all-ones afterwards)

        // Pass 2: sum of exp, plus select of the target logit (no scratch)
        float s  = 0.0f;
        float vt = 0.0f;
        #pragma unroll
        for (int c = 0; c < NCH; ++c) {
            s += __builtin_amdgcn_exp2f((v[c] - m) * LOG2E);   // v_exp_f32
            vt = (c == t) ? v[c] : vt;
        }

        // log-softmax at target class: logpt = (x_t - max) - ln(sum)
        float logpt = (vt - m) - __builtin_amdgcn_logf(s) * LN2;  // v_log_f32 = log2
        float pt    = __builtin_amdgcn_exp2f(logpt * LOG2E);
        float om    = 1.0f - pt;
        acc += -(om * om) * logpt;    // gamma = 2
    }

    // Deterministic block reduction in LDS
    __shared__ float sdata[NTHR];
    sdata[threadIdx.x] = acc;
    __syncthreads();
    #pragma unroll
    for (int off = NTHR / 2; off > 0; off >>= 1) {
        if (threadIdx.x < off) sdata[threadIdx.x] += sdata[threadIdx.x + off];
        __syncthreads();
    }
    if (threadIdx.x == 0) part[blockIdx.x] = sdata[0];
}

// Stage 2: one wave32 reduces the 1024 partials.
// Each lane sums 32 partials (fixed order), then one V_WMMA_F32_16X16X4_F32
// with B = all-ones collapses the 32 lane values: D[m][n] = a.x(lane m) +
// a.x(lane m+16) (K=1,3 slots are zero). Lane l then sums its 8 D VGPRs and
// one shfl_xor(16) merges the two halves. All products are *1.0 -> exact f32.
__global__ __launch_bounds__(32) void focal_finalize_kernel(
    const float* __restrict__ part,
    float*       __restrict__ out)
{
    const int lane = threadIdx.x;   // 0..31, EXEC all ones (WMMA requirement)

    float s = 0.0f;
    #pragma unroll 8
    for (int k = 0; k < NBLK / 32; ++k)
        s += part[lane + 32 * k];   // coalesced, fixed order -> deterministic

    v2f a;  a.x = s;    a.y = 0.0f;   // A[16x4]: lanes 0-15 hold K=0, lanes 16-31 hold K=2
    v2f bo; bo.x = 1.0f; bo.y = 1.0f; // B[4x16] = all ones (layout-independent)
    v8f c = {};
    v8f d = __builtin_amdgcn_wmma_f32_16x16x4_f32(
        /*neg_a=*/false, a, /*neg_b=*/false, bo,
        /*c_mod=*/(short)0, c, /*reuse_a=*/false, /*reuse_b=*/false);

    float half = d[0] + d[1] + d[2] + d[3] + d[4] + d[5] + d[6] + d[7];
    float total = half + __shfl_xor(half, 16, 32);

    if (lane == 0)
        out[0] = total * (1.0f / (float)NPIX);   // mean over all pixels
}

extern "C" void kernel_launch(void* const* d_in, const int* in_sizes, int n_in,
                              void* d_out, int out_size, void* d_ws, size_t ws_size,
                              hipStream_t stream) {
    const float* inp = (const float*)d_in[0];   // [8,21,512,512] f32
    const int*   tgt = (const int*)d_in[1];     // [8,512,512] int (harness int path)
    float* part = (float*)d_ws;                 // 1024 floats of scratch
    float* outp = (float*)d_out;                // scalar f32

    focal_partial_kernel<<<NBLK, NTHR, 0, stream>>>(inp, tgt, part);
    focal_finalize_kernel<<<1, 32, 0, stream>>>(part, outp);
}